// NutritionLoss_46291157516862
// MI455X (gfx1250) — compile-verified
//
#include <hip/hip_runtime.h>
#include <hip/hip_bf16.h>

// Problem constants (from reference)
#define BB 64
#define DD 7
#define MM 4
#define SS 16
#define NF 1000
#define NC 5
#define ELEMS (BB*DD*MM*SS)      // 28672
#define HSTRIDE 1024             // padded food axis (per batch row)
#define HMAT (BB*HSTRIDE)        // 65536 floats per histogram matrix
#define RMAT (BB*16)             // 1024 floats per result matrix (N padded to 16)
#define DPAD (NF*16)             // 16000 floats: zero-padded data [NF][16]

typedef float v2f __attribute__((ext_vector_type(2)));
typedef float v8f __attribute__((ext_vector_type(8)));

// --------------------------------------------- zero histograms + pad table
// h[0..2*HMAT)            : histograms, zeroed
// dpad[k*16+n]            : data[k][n] for n<5, else 0   (unconditional B loads)
__global__ void nl_prep_kernel(float* __restrict__ h,
                               float* __restrict__ dpad,
                               const float* __restrict__ data) {
    int i = blockIdx.x * blockDim.x + threadIdx.x;
    if (i < 2 * HMAT) h[i] = 0.0f;
    if (i < DPAD) {
        int k = i >> 4, n = i & 15;
        dpad[i] = (n < NC) ? data[k * NC + n] : 0.0f;
    }
}

// ------------------------------------------------- scatter amount histograms
__global__ void nl_hist_kernel(const float* __restrict__ y_pred,
                               const float* __restrict__ y,
                               float* __restrict__ h) {
    int idx = blockIdx.x * blockDim.x + threadIdx.x;
    if (idx >= ELEMS) return;
    int b = idx / (DD * MM * SS);

    float tid_f = y[idx * 2 + 0];
    float tamt  = y[idx * 2 + 1];
    float pid_f = y_pred[idx * 2 + 0];
    float pamt  = y_pred[idx * 2 + 1];

    // jnp.round == rintf (round-half-even); ids never land on .5 exactly
    int tid = (int)rintf(tid_f);
    int pid = (int)rintf(pid_f);
    tid = tid < 0 ? 0 : (tid >= NF ? NF - 1 : tid);
    pid = pid < 0 ? 0 : (pid >= NF ? NF - 1 : pid);

    atomicAdd(&h[b * HSTRIDE + tid], tamt);
    atomicAdd(&h[HMAT + b * HSTRIDE + pid], pamt);
}

// ------------------------------------------------- R = H @ dataPad via f32 WMMA
// grid = 8 blocks ({gold,pred} x 4 M-tiles), block = 32 (one wave, EXEC all-1s)
__global__ void nl_gemm_kernel(const float* __restrict__ h,
                               const float* __restrict__ dpad,
                               float* __restrict__ r) {
    const int which = blockIdx.x & 1;
    const int tm    = blockIdx.x >> 1;          // batch tile: rows tm*16..tm*16+15
    const float* H  = h + which * HMAT + tm * 16 * HSTRIDE;

    const int lane = threadIdx.x;               // wave32
    const int half = lane >> 4;                 // 0: K=0,1  1: K=2,3
    const int mn   = lane & 15;                 // M for A, N for B

    __builtin_prefetch(dpad, 0, 3);             // global_prefetch_b8 (64KB table)

    // Per-lane base pointers; both A and B advance by 4 K per step.
    const float* ap = H    + mn * HSTRIDE + half * 2;   // A: row mn, K = half*2 (+1)
    const float* bp = dpad + half * 2 * 16 + mn;        // B: K = half*2 (+1), col mn

    v8f acc = {0.f, 0.f, 0.f, 0.f, 0.f, 0.f, 0.f, 0.f};

    // K = 1000 = 250 steps of K=4; all loads unconditional b64/b32, no branches.
    for (int k0 = 0; k0 < NF; k0 += 4) {
        v2f a = *(const v2f*)ap;                // contiguous: global_load_b64
        v2f bv;
        bv.x = bp[0];
        bv.y = bp[16];
        acc = __builtin_amdgcn_wmma_f32_16x16x4_f32(
            false, a, false, bv, (short)0, acc, false, false);
        ap += 4;
        bp += 4 * 16;
    }

    // D layout: VGPR rr -> M = rr + half*8, N = mn
    float* out = r + which * RMAT + tm * 16 * 16;
    #pragma unroll
    for (int rr = 0; rr < 8; ++rr) {
        int m = rr + half * 8;
        out[m * 16 + mn] = acc[rr];
    }
}

// ------------------------------------------------- final L1 reduction
// loss = 5 * sum_c mean_b |pred - gold| / 700
__global__ void nl_reduce_kernel(const float* __restrict__ r,
                                 float* __restrict__ out) {
    __shared__ float sbuf[256];
    int t = threadIdx.x;
    float acc = 0.0f;
    for (int p = t; p < BB * NC; p += 256) {
        int b = p / NC;
        int c = p % NC;
        float g = r[b * 16 + c];
        float q = r[RMAT + b * 16 + c];
        acc += fabsf(q - g);
    }
    sbuf[t] = acc;
    __syncthreads();
    for (int s = 128; s > 0; s >>= 1) {
        if (t < s) sbuf[t] += sbuf[t + s];
        __syncthreads();
    }
    if (t == 0) out[0] = sbuf[0] * (5.0f / (64.0f * 700.0f));
}

extern "C" void kernel_launch(void* const* d_in, const int* in_sizes, int n_in,
                              void* d_out, int out_size, void* d_ws, size_t ws_size,
                              hipStream_t stream) {
    const float* y_pred = (const float*)d_in[0];   // [B,D,M,S,2]
    const float* y      = (const float*)d_in[1];   // [B,D,M,S,2]
    const float* data   = (const float*)d_in[2];   // [NF,NC]
    float* out = (float*)d_out;

    float* h    = (float*)d_ws;                    // 2 * HMAT floats
    float* r    = h + 2 * HMAT;                    // 2 * RMAT floats
    float* dpad = r + 2 * RMAT;                    // DPAD floats

    // 1) zero histograms + build zero-padded B matrix
    {
        int n = 2 * HMAT;                          // 131072 >= DPAD
        nl_prep_kernel<<<(n + 255) / 256, 256, 0, stream>>>(h, dpad, data);
    }
    // 2) scatter gold/pred weighted-amount histograms
    nl_hist_kernel<<<(ELEMS + 255) / 256, 256, 0, stream>>>(y_pred, y, h);
    // 3) R = H @ dataPad  (fp32 WMMA, 8 waves, 250 wmma each, branch-free loop)
    nl_gemm_kernel<<<8, 32, 0, stream>>>(h, dpad, r);
    // 4) L1 / mean / scale
    nl_reduce_kernel<<<1, 256, 0, stream>>>(r, out);
}